// RIME_76338748719830
// MI455X (gfx1250) — compile-verified
//
#include <hip/hip_runtime.h>
#include <math.h>

#define NANTS  32
#define NBEAM  4
#define NTIMES 8
#define NFREQS 32
#define NPIX   2048
#define NBLS   64
#define INV_C  (1.0f / 299792458.0f)

typedef float v2f __attribute__((ext_vector_type(2)));
typedef float v8f __attribute__((ext_vector_type(8)));

// ---------------------------------------------------------------------------
// Kernel 1: tau[b][t][s] = dot(bl_vecs[b], s_topo[t,:,s]) / c
// Genuine GEMM (64x3)x(3x(8*2048)), K=3 padded to 4 -> V_WMMA_F32_16X16X4_F32.
// Grid: (NPIX/16, NTIMES), block 128 (4 waves); wave w owns baseline tile w.
// ---------------------------------------------------------------------------
__global__ void tau_wmma_kernel(const float* __restrict__ bl_vecs,   // (64,3)
                                const float* __restrict__ s_topo,    // (8,3,2048)
                                float* __restrict__ tau)             // (64,8,2048)
{
    const int lane = threadIdx.x & 31;
    const int wave = threadIdx.x >> 5;          // baseline tile 0..3
    const int half = lane >> 4;                 // 0: lanes 0-15, 1: lanes 16-31
    const int l16  = lane & 15;
    const int t    = blockIdx.y;
    const int pix0 = blockIdx.x * 16;
    const int m    = wave * 16 + l16;           // A row (baseline)
    const int n    = pix0 + l16;                // B col (pixel)

    // A (16x4), 32-bit layout: VGPR0 = K0 (lanes0-15) / K2 (lanes16-31),
    //                          VGPR1 = K1 (lanes0-15) / K3 (lanes16-31).
    v2f a;
    a.x = half ? bl_vecs[m * 3 + 2] : bl_vecs[m * 3 + 0];   // z : x
    a.y = half ? 0.0f               : bl_vecs[m * 3 + 1];   // 0 : y
    // B (4x16), mirrored layout: rows are s_topo coordinate planes.
    v2f bm;
    bm.x = half ? s_topo[(t * 3 + 2) * NPIX + n] : s_topo[(t * 3 + 0) * NPIX + n];
    bm.y = half ? 0.0f                           : s_topo[(t * 3 + 1) * NPIX + n];

    v8f c = {0.f, 0.f, 0.f, 0.f, 0.f, 0.f, 0.f, 0.f};
    v8f d = __builtin_amdgcn_wmma_f32_16x16x4_f32(false, a, false, bm,
                                                  (short)0, c, false, false);

    // D 16x16 f32 layout: VGPR r -> (M=r, N=lane) lanes0-15, (M=r+8) lanes16-31.
    // Reciprocal-multiply instead of IEEE divide: <=1 ulp on tau, avoids the
    // 12-instruction v_div_scale/v_div_fmas expansion per element.
#pragma unroll
    for (int r = 0; r < 8; ++r) {
        int b = wave * 16 + r + half * 8;
        tau[(b * NTIMES + t) * NPIX + (pix0 + l16)] = d[r] * INV_C;
    }
}

// ---------------------------------------------------------------------------
// Kernel 2: vis[b,t,f] = sum_s sky*(b1*conj(b2)) * exp(i*2pi*f*tau)
// One block per (b,t); wave w (of 8) owns freqs [4w,4w+4), all 2048 pixels.
// Cross-lane reduce with 5 shfl_xor steps (wave32), lane 0 stores complex.
// ---------------------------------------------------------------------------
template <bool USE_WS>
__global__ __launch_bounds__(256) void vis_kernel(
    const float* __restrict__ beams,    // (4,8,32,2048) complex64 interleaved
    const float* __restrict__ sky,      // (32,2048)
    const float* __restrict__ freqs,    // (32)
    const int*   __restrict__ b1idx,    // (64)
    const int*   __restrict__ b2idx,    // (64)
    const float* __restrict__ tau_ws,   // (64,8,2048) when USE_WS
    const float* __restrict__ bl_vecs,  // (64,3)      when !USE_WS
    const float* __restrict__ s_topo,   // (8,3,2048)  when !USE_WS
    float* __restrict__ out)            // (64,8,32) complex64 interleaved
{
    const int b    = blockIdx.x >> 3;
    const int t    = blockIdx.x & 7;
    const int wave = threadIdx.x >> 5;      // 0..7
    const int lane = threadIdx.x & 31;
    const int fbase = wave * 4;

    const int m1 = b1idx[b];
    const int m2 = b2idx[b];

    float wf[4];
#pragma unroll
    for (int ff = 0; ff < 4; ++ff)
        wf[ff] = 6.28318530717958647692f * freqs[fbase + ff];

    float blx = 0.f, bly = 0.f, blz = 0.f;
    if (!USE_WS) {
        blx = bl_vecs[b * 3 + 0];
        bly = bl_vecs[b * 3 + 1];
        blz = bl_vecs[b * 3 + 2];
    }
    const float* taub = USE_WS ? (tau_ws + (size_t)(b * NTIMES + t) * NPIX) : nullptr;

    const float2* bm1 = (const float2*)beams + (size_t)(m1 * NTIMES + t) * NFREQS * NPIX;
    const float2* bm2 = (const float2*)beams + (size_t)(m2 * NTIMES + t) * NFREQS * NPIX;

    float accr[4] = {0.f, 0.f, 0.f, 0.f};
    float acci[4] = {0.f, 0.f, 0.f, 0.f};

    for (int sb = 0; sb < NPIX / 32; ++sb) {
        const int s = sb * 32 + lane;
        float ta;
        if (USE_WS) {
            ta = taub[s];
        } else {
            float sx = s_topo[(t * 3 + 0) * NPIX + s];
            float sy = s_topo[(t * 3 + 1) * NPIX + s];
            float sz = s_topo[(t * 3 + 2) * NPIX + s];
            ta = (blx * sx + bly * sy + blz * sz) * INV_C;
        }
#pragma unroll
        for (int ff = 0; ff < 4; ++ff) {
            const int f = fbase + ff;
            float ph = wf[ff] * ta;
            float sn, cs;
            sincosf(ph, &sn, &cs);   // accurate path: |phase| ~ 2e3 rad > 256*pi,
                                     // outside native v_sin_f32's valid range
            float  sk = sky[f * NPIX + s];
            float2 v1 = bm1[(size_t)f * NPIX + s];
            float2 v2 = bm2[(size_t)f * NPIX + s];
            // b1 * conj(b2)
            float pr = sk * fmaf(v1.x, v2.x,  v1.y * v2.y);
            float pi = sk * fmaf(v1.y, v2.x, -v1.x * v2.y);
            // * fringe, accumulate
            accr[ff] += fmaf(pr, cs, -pi * sn);
            acci[ff] += fmaf(pr, sn,  pi * cs);
        }
    }

#pragma unroll
    for (int ff = 0; ff < 4; ++ff) {
        float r = accr[ff], i = acci[ff];
        for (int off = 16; off > 0; off >>= 1) {
            r += __shfl_xor(r, off, 32);
            i += __shfl_xor(i, off, 32);
        }
        if (lane == 0) {
            const int f = fbase + ff;
            size_t o = ((size_t)(b * NTIMES + t) * NFREQS + f) * 2;
            out[o + 0] = r;
            out[o + 1] = i;
        }
    }
}

// ---------------------------------------------------------------------------
extern "C" void kernel_launch(void* const* d_in, const int* in_sizes, int n_in,
                              void* d_out, int out_size, void* d_ws, size_t ws_size,
                              hipStream_t stream) {
    (void)in_sizes; (void)n_in; (void)out_size;
    const float* beams = (const float*)d_in[0];   // ant_beams (complex64)
    const float* sky   = (const float*)d_in[1];   // sky
    const float* blv   = (const float*)d_in[2];   // bl_vecs
    const float* stp   = (const float*)d_in[3];   // s_topo
    const float* frq   = (const float*)d_in[4];   // freqs
    const int*   b1    = (const int*)d_in[5];     // beam1_idx
    const int*   b2    = (const int*)d_in[6];     // beam2_idx
    float* out = (float*)d_out;

    const size_t tau_bytes = (size_t)NBLS * NTIMES * NPIX * sizeof(float);
    if (ws_size >= tau_bytes) {
        float* tau = (float*)d_ws;
        tau_wmma_kernel<<<dim3(NPIX / 16, NTIMES), 128, 0, stream>>>(blv, stp, tau);
        vis_kernel<true><<<NBLS * NTIMES, 256, 0, stream>>>(
            beams, sky, frq, b1, b2, tau, nullptr, nullptr, out);
    } else {
        vis_kernel<false><<<NBLS * NTIMES, 256, 0, stream>>>(
            beams, sky, frq, b1, b2, nullptr, blv, stp, out);
    }
}